// Encoder_76733885710747
// MI455X (gfx1250) — compile-verified
//
#include <hip/hip_runtime.h>
#include <hip/hip_bf16.h>

typedef __attribute__((ext_vector_type(2))) float v2f;
typedef __attribute__((ext_vector_type(8))) float v8f;

constexpr int BB  = 32;
constexpr int SS  = 2048;
constexpr int DD  = 64;
constexpr int DK  = 24;
constexpr int DV  = 32;
constexpr int LDQ = 32;        // padded row stride (floats) for Q,K
constexpr int NT  = SS / 16;   // 128 16-wide tiles along S

#define WMMA_F32X4(Am, Bm, Cm) \
  __builtin_amdgcn_wmma_f32_16x16x4_f32(false, (Am), false, (Bm), (short)0, (Cm), false, false)

// ---------------------------------------------------------------------------
// Kernel 1: q = x@Wq, k = x@Wk, v = x@Wv.  One thread per (b,s) row. ~0.7 GFLOP.
// Q,K stored with row stride 32 floats (128B rows, 8B-aligned v2f WMMA loads).
// ---------------------------------------------------------------------------
__global__ void __launch_bounds__(256)
qkv_proj_kernel(const float* __restrict__ x, const float* __restrict__ Wq,
                const float* __restrict__ Wk, const float* __restrict__ Wv,
                float* __restrict__ Qp, float* __restrict__ Kp, float* __restrict__ Vm)
{
    int row = blockIdx.x * blockDim.x + threadIdx.x;
    if (row >= BB * SS) return;
    const float* xp = x + (size_t)row * DD;

    float accq[DK], acck[DK], accv[DV];
    #pragma unroll
    for (int j = 0; j < DK; ++j) { accq[j] = 0.f; acck[j] = 0.f; }
    #pragma unroll
    for (int j = 0; j < DV; ++j) accv[j] = 0.f;

    for (int d = 0; d < DD; ++d) {
        float xd = xp[d];
        #pragma unroll
        for (int j = 0; j < DK; ++j) {
            accq[j] = fmaf(xd, Wq[d * DK + j], accq[j]);
            acck[j] = fmaf(xd, Wk[d * DK + j], acck[j]);
        }
        #pragma unroll
        for (int j = 0; j < DV; ++j)
            accv[j] = fmaf(xd, Wv[d * DV + j], accv[j]);
    }
    float* qo = Qp + (size_t)row * LDQ;
    float* ko = Kp + (size_t)row * LDQ;
    float* vo = Vm + (size_t)row * DV;
    #pragma unroll
    for (int j = 0; j < DK; ++j) { qo[j] = accq[j]; ko[j] = acck[j]; }
    #pragma unroll
    for (int j = 0; j < DV; ++j) vo[j] = accv[j];
}

// ---------------------------------------------------------------------------
// Kernel 2: per-key-column softmax stats over the query axis.
// One wave per (batch, 16-wide t-tile). B-tile (K^T) is loop-invariant; loop
// streams 128 q-tiles, each scored with 6 chained V_WMMA_F32_16X16X4_F32
// (K = 24 = 6x4). Online max/sum per column via in-lane reduce + shfl_xor(16).
// Writes m[t] and 1/Z[t].
// ---------------------------------------------------------------------------
__global__ void __launch_bounds__(32)
col_softmax_stats_kernel(const float* __restrict__ Qp, const float* __restrict__ Kp,
                         float* __restrict__ mArr, float* __restrict__ zinvArr)
{
    int wave = blockIdx.x;
    int bb = wave / NT;
    int tt = wave - bb * NT;
    int lane = threadIdx.x & 31;
    int col  = lane & 15;        // N index of D tile / row index for A,B loads
    int hl   = lane >> 4;        // lane half -> K sub-pair

    // B tile: B[kd][n] = K[tbase+n][kd], kd = 4c + 2*hl + {0,1}
    const float* kp = Kp + ((size_t)(bb * SS) + (size_t)tt * 16 + col) * LDQ + 2 * hl;
    v2f b0 = *(const v2f*)(kp +  0);
    v2f b1 = *(const v2f*)(kp +  4);
    v2f b2 = *(const v2f*)(kp +  8);
    v2f b3 = *(const v2f*)(kp + 12);
    v2f b4 = *(const v2f*)(kp + 16);
    v2f b5 = *(const v2f*)(kp + 20);

    const float* qbase = Qp + ((size_t)(bb * SS) + col) * LDQ + 2 * hl;
    float m_run = -3.0e38f;
    float z_run = 0.f;

    for (int qt = 0; qt < NT; ++qt) {
        const float* qp = qbase + (size_t)qt * 16 * LDQ;
        __builtin_prefetch(qp + 16 * LDQ, 0, 1);   // global_prefetch next q tile
        v2f a0 = *(const v2f*)(qp +  0);
        v2f a1 = *(const v2f*)(qp +  4);
        v2f a2 = *(const v2f*)(qp +  8);
        v2f a3 = *(const v2f*)(qp + 12);
        v2f a4 = *(const v2f*)(qp + 16);
        v2f a5 = *(const v2f*)(qp + 20);

        v8f s = {};
        s = WMMA_F32X4(a0, b0, s);
        s = WMMA_F32X4(a1, b1, s);
        s = WMMA_F32X4(a2, b2, s);
        s = WMMA_F32X4(a3, b3, s);
        s = WMMA_F32X4(a4, b4, s);
        s = WMMA_F32X4(a5, b5, s);

        // column (over q) online softmax update; lane holds 8 rows of its column
        float tmax = fmaxf(fmaxf(fmaxf(s[0], s[1]), fmaxf(s[2], s[3])),
                           fmaxf(fmaxf(s[4], s[5]), fmaxf(s[6], s[7])));
        tmax = fmaxf(tmax, __shfl_xor(tmax, 16, 32));
        float m_new = fmaxf(m_run, tmax);
        float p = 0.f;
        #pragma unroll
        for (int i = 0; i < 8; ++i) p += __expf(s[i] - m_new);
        p += __shfl_xor(p, 16, 32);
        z_run = fmaf(z_run, __expf(m_run - m_new), p);
        m_run = m_new;
    }
    if (hl == 0) {
        mArr[bb * SS + tt * 16 + col]    = m_run;
        zinvArr[bb * SS + tt * 16 + col] = 1.f / z_run;
    }
}

// ---------------------------------------------------------------------------
// Kernel 3: fold 1/Z into v (in place; v is recomputed each launch).
// ---------------------------------------------------------------------------
__global__ void __launch_bounds__(256)
scale_v_kernel(float* __restrict__ Vm, const float* __restrict__ zinvArr)
{
    int i = blockIdx.x * blockDim.x + threadIdx.x;
    if (i < BB * SS * DV) Vm[i] *= zinvArr[i / DV];
}

// ---------------------------------------------------------------------------
// Kernel 4: h1[q,:] = sum_t exp(s_qt - m[t]) * vz[t,:].
// One wave per (batch, 16-wide q-tile). A (Q tile) is loop-invariant; loop
// streams 128 t-tiles: 6 WMMA for scores, exp to LDS (D-layout -> A-layout
// transpose, stride 20 = conflict-free), then 8 WMMA (4 K-chunks x 2 N-tiles)
// accumulating against vz loaded from global.
// ---------------------------------------------------------------------------
__global__ void __launch_bounds__(32)
attn_h1_kernel(const float* __restrict__ Qp, const float* __restrict__ Kp,
               const float* __restrict__ VZ, const float* __restrict__ mArr,
               float* __restrict__ h1)
{
    __shared__ float ldsE[16 * 20];
    int wave = blockIdx.x;
    int bb = wave / NT;
    int qt = wave - bb * NT;
    int lane = threadIdx.x & 31;
    int col  = lane & 15;
    int hl   = lane >> 4;

    const float* qp = Qp + ((size_t)(bb * SS) + (size_t)qt * 16 + col) * LDQ + 2 * hl;
    v2f a0 = *(const v2f*)(qp +  0);
    v2f a1 = *(const v2f*)(qp +  4);
    v2f a2 = *(const v2f*)(qp +  8);
    v2f a3 = *(const v2f*)(qp + 12);
    v2f a4 = *(const v2f*)(qp + 16);
    v2f a5 = *(const v2f*)(qp + 20);

    v8f acc0 = {};   // dv 0..15
    v8f acc1 = {};   // dv 16..31

    const float* kbase = Kp + ((size_t)(bb * SS) + col) * LDQ + 2 * hl;
    const float* vzb   = VZ + (size_t)(bb * SS) * DV;
    const float* mrow  = mArr + bb * SS;

    for (int tt = 0; tt < NT; ++tt) {
        const float* kp = kbase + (size_t)tt * 16 * LDQ;
        __builtin_prefetch(kp + 16 * LDQ, 0, 1);   // global_prefetch next k tile
        v2f b0 = *(const v2f*)(kp +  0);
        v2f b1 = *(const v2f*)(kp +  4);
        v2f b2 = *(const v2f*)(kp +  8);
        v2f b3 = *(const v2f*)(kp + 12);
        v2f b4 = *(const v2f*)(kp + 16);
        v2f b5 = *(const v2f*)(kp + 20);

        v8f s = {};
        s = WMMA_F32X4(a0, b0, s);
        s = WMMA_F32X4(a1, b1, s);
        s = WMMA_F32X4(a2, b2, s);
        s = WMMA_F32X4(a3, b3, s);
        s = WMMA_F32X4(a4, b4, s);
        s = WMMA_F32X4(a5, b5, s);

        // E = exp(s - m[t]) written in D layout: row q_local = r+8*hl, col t_local
        float mcol = mrow[tt * 16 + col];
        #pragma unroll
        for (int r = 0; r < 8; ++r)
            ldsE[(r + 8 * hl) * 20 + col] = __expf(s[r] - mcol);
        // single wave owns this LDS tile: DS ops are in-order, compiler inserts
        // s_wait_dscnt before the dependent reads below — no barrier needed.

        const float* vt = vzb + (size_t)(tt * 16) * DV;
        #pragma unroll
        for (int c = 0; c < 4; ++c) {
            // A chunk: A[m][kd] = E[q_local = col][t_local = 4c + 2*hl + {0,1}]
            v2f ae = *(const v2f*)&ldsE[col * 20 + 4 * c + 2 * hl];
            const float* vp = vt + (size_t)(4 * c + 2 * hl) * DV;
            v2f bv0, bv1;
            bv0.x = vp[col];           bv0.y = vp[DV + col];        // rows kd, kd+1
            bv1.x = vp[16 + col];      bv1.y = vp[DV + 16 + col];
            acc0 = WMMA_F32X4(ae, bv0, acc0);
            acc1 = WMMA_F32X4(ae, bv1, acc1);
        }
    }

    #pragma unroll
    for (int r = 0; r < 8; ++r) {
        float* hp = h1 + ((size_t)(bb * SS) + (size_t)qt * 16 + r + 8 * hl) * DV;
        hp[col]      = acc0[r];
        hp[col + 16] = acc1[r];
    }
}

// ---------------------------------------------------------------------------
// Kernel 5: out = h1 @ Wh  (32 -> 64). One thread per (b,s) row. ~0.27 GFLOP.
// ---------------------------------------------------------------------------
__global__ void __launch_bounds__(256)
out_proj_kernel(const float* __restrict__ h1, const float* __restrict__ Wh,
                float* __restrict__ out)
{
    int row = blockIdx.x * blockDim.x + threadIdx.x;
    if (row >= BB * SS) return;
    const float* hp = h1 + (size_t)row * DV;
    float acc[DD];
    #pragma unroll
    for (int d = 0; d < DD; ++d) acc[d] = 0.f;
    for (int v = 0; v < DV; ++v) {
        float hv = hp[v];
        #pragma unroll
        for (int d = 0; d < DD; ++d)
            acc[d] = fmaf(hv, Wh[v * DD + d], acc[d]);
    }
    float* op = out + (size_t)row * DD;
    #pragma unroll
    for (int d = 0; d < DD; ++d) op[d] = acc[d];
}

// ---------------------------------------------------------------------------
extern "C" void kernel_launch(void* const* d_in, const int* in_sizes, int n_in,
                              void* d_out, int out_size, void* d_ws, size_t ws_size,
                              hipStream_t stream)
{
    const float* x  = (const float*)d_in[0];
    const float* Wq = (const float*)d_in[1];
    const float* Wk = (const float*)d_in[2];
    const float* Wv = (const float*)d_in[3];
    const float* Wh = (const float*)d_in[4];
    float* out = (float*)d_out;

    // workspace carve-up (floats); total ~32.5 MB
    float* ws = (float*)d_ws;
    size_t o = 0;
    float* Qp = ws + o; o += (size_t)BB * SS * LDQ;   // 8 MB
    float* Kp = ws + o; o += (size_t)BB * SS * LDQ;   // 8 MB
    float* Vm = ws + o; o += (size_t)BB * SS * DV;    // 8 MB (becomes vz in place)
    float* mA = ws + o; o += (size_t)BB * SS;         // 256 KB
    float* zI = ws + o; o += (size_t)BB * SS;         // 256 KB
    float* h1 = ws + o; o += (size_t)BB * SS * DV;    // 8 MB

    const int rows = BB * SS;
    qkv_proj_kernel<<<(rows + 255) / 256, 256, 0, stream>>>(x, Wq, Wk, Wv, Qp, Kp, Vm);
    col_softmax_stats_kernel<<<BB * NT, 32, 0, stream>>>(Qp, Kp, mA, zI);
    const int nv = BB * SS * DV;
    scale_v_kernel<<<(nv + 255) / 256, 256, 0, stream>>>(Vm, zI);
    attn_h1_kernel<<<BB * NT, 32, 0, stream>>>(Qp, Kp, Vm, mA, h1);
    out_proj_kernel<<<(rows + 255) / 256, 256, 0, stream>>>(h1, Wh, out);
}